// Quantize_34540126994608
// MI455X (gfx1250) — compile-verified
//
#include <hip/hip_runtime.h>

typedef __attribute__((ext_vector_type(2))) float v2f;
typedef __attribute__((ext_vector_type(8))) float v8f;

#define VQ_DIM     64
#define VQ_NEMBED  1024
#define VQ_NROWS   65536                 // 16*64*64
#define VQ_QELEMS  (VQ_NROWS * VQ_DIM)   // 4194304
#define VQ_PANELW  64                    // codes per LDS panel
#define VQ_NPANEL  (VQ_NEMBED / VQ_PANELW)  // 16

// --- CDNA5 async global->LDS path, guarded so the file always compiles -----
#if defined(__gfx1250__) && __has_builtin(__builtin_amdgcn_global_load_async_to_lds_b32)
#define VQ_ASYNC_LDS 1
#else
#define VQ_ASYNC_LDS 0
#endif

#if VQ_ASYNC_LDS
typedef __attribute__((address_space(1))) int as1_i32;   // global
typedef __attribute__((address_space(3))) int as3_i32;   // LDS
#endif

__device__ __forceinline__ void vq_async_wait0() {
#if VQ_ASYNC_LDS
#if __has_builtin(__builtin_amdgcn_s_wait_asynccnt)
    __builtin_amdgcn_s_wait_asynccnt(0);
#else
    asm volatile("s_wait_asynccnt 0" ::: "memory");
#endif
#endif
}

// ---------------------------------------------------------------------------
// Prelude: ||e_n||^2 -> ws, zero the diff/zeros output slots.
// ---------------------------------------------------------------------------
__global__ void vq_enorm_kernel(const float* __restrict__ embed,
                                float* __restrict__ enorm,
                                float* __restrict__ scalars /* [diff, zeros] */) {
    int n = blockIdx.x * blockDim.x + threadIdx.x;
    if (n < VQ_NEMBED) {
        float s = 0.0f;
        #pragma unroll
        for (int k = 0; k < VQ_DIM; ++k) {
            float e = embed[k * VQ_NEMBED + n];
            s += e * e;
        }
        enorm[n] = s;
    }
    if (blockIdx.x == 0 && threadIdx.x == 0) {
        scalars[0] = 0.0f;
        scalars[1] = 0.0f;
    }
}

// ---------------------------------------------------------------------------
// Fused GEMM(argmin) + gather + MSE. Embed panels live in LDS in WMMA
// B-fragment pair order: pair(kb, n, half) = (e[k0][n], e[k0+1][n]) at pair
// index kb*128 + n*2 + half  (k0 = kb*4 + half*2). Each lane's B operand is
// one ds_load_b64, conflict-free across all 64 banks.
// ---------------------------------------------------------------------------
__global__ __launch_bounds__(256) void vq_main_kernel(
        const float* __restrict__ x,
        const float* __restrict__ embed,
        const float* __restrict__ enorm,
        float* __restrict__ qout,
        float* __restrict__ diff_out,
        float* __restrict__ ind_out) {
    __shared__ float sB[2][VQ_DIM * VQ_PANELW];   // 2 x 16KB, pair-interleaved

    const int tid     = threadIdx.x;
    const int lane    = tid & 31;
    const int wave    = tid >> 5;
    const int half    = lane >> 4;     // 0: lanes 0-15, 1: lanes 16-31
    const int col     = lane & 15;
    const int lpair   = col * 2 + half;        // lane's pair slot within a row
    const int tileid  = blockIdx.x * 8 + wave;
    const int rowbase = tileid * 16;
    const int row     = rowbase + col;

    // Cooperative interleaving copy: thread owns e[ck][cn0..cn0+15].
    const int ck    = tid >> 2;        // k row 0..63
    const int cn0   = (tid & 3) * 16;  // col offset 0/16/32/48
    const int ckb   = ck >> 2;
    const int chalf = (ck >> 1) & 1;
    const int ckl   = ck & 1;
    // LDS float index of element (ck, n): ckb*256 + n*4 + chalf*2 + ckl
    const int lbase = ckb * 256 + cn0 * 4 + chalf * 2 + ckl;

    auto load_panel = [&](int p, int buf) {
        const float* src = embed + (size_t)ck * VQ_NEMBED + p * VQ_PANELW + cn0;
        float*       dst = &sB[buf][lbase];
        #pragma unroll
        for (int j = 0; j < 16; ++j) {
#if VQ_ASYNC_LDS
            __builtin_amdgcn_global_load_async_to_lds_b32(
                (as1_i32*)(const_cast<float*>(src) + j),
                (as3_i32*)(dst + j * 4),
                /*offset=*/0, /*cpol=*/0);
#else
            dst[j * 4] = src[j];
#endif
        }
    };

    // A fragments: 16x4 f32 tiles. lanes 0-15: K={0,1}; lanes 16-31: K={2,3}.
    v2f a[16];
    const float* xrow = x + (size_t)row * VQ_DIM + half * 2;
    #pragma unroll
    for (int kb = 0; kb < 16; ++kb) {
        a[kb].x = xrow[kb * 4 + 0];
        a[kb].y = xrow[kb * 4 + 1];
    }

    float minv[8];
    int   mini[8];
    #pragma unroll
    for (int r = 0; r < 8; ++r) { minv[r] = 3.4e38f; mini[r] = 0; }

    load_panel(0, 0);
    for (int p = 0; p < VQ_NPANEL; ++p) {
        const int buf = p & 1;
        vq_async_wait0();      // panel p landed (this thread's async ops)
        __syncthreads();       // all threads: panel p visible; prev buf free
        if (p + 1 < VQ_NPANEL) load_panel(p + 1, buf ^ 1);

        const v2f* bp = (const v2f*)&sB[buf][0];
        #pragma unroll
        for (int tt = 0; tt < VQ_PANELW / 16; ++tt) {
            const int ncol = p * VQ_PANELW + tt * 16 + col;  // global code idx
            v8f c = {};
            #pragma unroll
            for (int kb = 0; kb < 16; ++kb) {
                // single ds_load_b64 into an aligned VGPR pair
                v2f b = bp[kb * 128 + tt * 32 + lpair];
                c = __builtin_amdgcn_wmma_f32_16x16x4_f32(
                        false, a[kb], false, b, (short)0, c, false, false);
            }
            const float en = enorm[ncol];
            #pragma unroll
            for (int r = 0; r < 8; ++r) {
                float d = en - 2.0f * c[r];
                if (d < minv[r]) { minv[r] = d; mini[r] = ncol; }
            }
        }
    }

    // Cross-lane argmin within 16-lane halves; ties -> lowest index.
    #pragma unroll
    for (int r = 0; r < 8; ++r) {
        #pragma unroll
        for (int m = 8; m >= 1; m >>= 1) {
            float ov = __shfl_xor(minv[r], m, 32);
            int   oi = __shfl_xor(mini[r], m, 32);
            if (ov < minv[r] || (ov == minv[r] && oi < mini[r])) {
                minv[r] = ov; mini[r] = oi;
            }
        }
    }

    // Gather winning codes, write quantize + indices, accumulate MSE.
    float dsum = 0.0f;
    #pragma unroll
    for (int m = 0; m < 16; ++m) {
        const int r       = m & 7;
        const int srclane = (m < 8) ? 0 : 16;
        const int bi      = __shfl(mini[r], srclane, 32);
        const size_t orow = (size_t)(rowbase + m) * VQ_DIM;
        #pragma unroll
        for (int kk = 0; kk < 2; ++kk) {
            const int k = kk * 32 + lane;
            float q   = embed[(size_t)k * VQ_NEMBED + bi];
            float xv  = x[orow + k];
            float dlt = q - xv;
            dsum += dlt * dlt;
            qout[orow + k] = q;
        }
        if (lane == srclane) ind_out[rowbase + m] = (float)bi;
    }

    #pragma unroll
    for (int m = 16; m >= 1; m >>= 1) dsum += __shfl_xor(dsum, m, 32);
    if (lane == 0) atomicAdd(diff_out, dsum * (1.0f / (float)VQ_QELEMS));
}

// ---------------------------------------------------------------------------
extern "C" void kernel_launch(void* const* d_in, const int* in_sizes, int n_in,
                              void* d_out, int out_size, void* d_ws, size_t ws_size,
                              hipStream_t stream) {
    const float* x     = (const float*)d_in[0];   // [16,64,64,64] f32
    const float* embed = (const float*)d_in[1];   // [64,1024] f32

    float* out     = (float*)d_out;
    float* qout    = out;                         // quantize: 4194304 f32
    float* scalars = out + VQ_QELEMS;             // [diff, zeros]
    float* ind_out = out + VQ_QELEMS + 2;         // embed_ind: 65536
    float* enorm   = (float*)d_ws;                // 1024 f32 scratch

    vq_enorm_kernel<<<4, 256, 0, stream>>>(embed, enorm, scalars);
    vq_main_kernel<<<VQ_NROWS / 16 / 8, 256, 0, stream>>>(
        x, embed, enorm, qout, scalars, ind_out);
}